// ConvHybridFilter_65481071402487
// MI455X (gfx1250) — compile-verified
//
#include <hip/hip_runtime.h>
#include <hip/hip_bf16.h>

// D = A*B + C on the f32 WMMA pipe: A is 16x4 f32 (2 VGPRs/lane),
// B is 4x16 f32 (2 VGPRs/lane), C/D 16x16 f32 (8 VGPRs/lane).
typedef float v2f __attribute__((ext_vector_type(2)));
typedef float v8f __attribute__((ext_vector_type(8)));

#define IMG_W       64
#define IMG_H       64
#define OUT_W       61
#define OUT_H       61
#define BATCH       256
#define N_OUT_TOTAL (BATCH * OUT_H * OUT_W)   // 952,576 conv outputs

__global__ void zero_acc_kernel(float* acc) {
    if (threadIdx.x == 0) acc[0] = 0.0f;
}

// One wave per (batch image, output row, 16-wide column chunk).
// Conv 4x4 as im2col GEMM via 4 chained V_WMMA_F32_16X16X4_F32 (full f32
// precision, RNE — matches the reference numerically up to FMA ordering).
__global__ __launch_bounds__(256) void conv_sigmoid_reduce_wmma(
    const float* __restrict__ data,   // (256,1,64,64)
    const float* __restrict__ w,      // (1,1,4,4) -> w[16]
    const float* __restrict__ bptr,   // (1,)
    float* __restrict__ acc)          // global sum of sigmoid(logits)
{
    __shared__ float wsum[8];

    const int lane  = threadIdx.x & 31;
    const int wv    = threadIdx.x >> 5;          // wave in block: 0..7
    const int gw    = blockIdx.x * 8 + wv;       // global wave id
    const int chunk = gw & 3;                    // 0..3 (16 cols each)
    const int t     = gw >> 2;
    const int row   = t % OUT_H;                 // output row 0..60
    const int b     = t / OUT_H;                 // image 0..255
    const int j0    = chunk * 16;
    const int h     = lane >> 4;                 // K-half select (0/1)
    const int n     = lane & 15;                 // output column in chunk
    const int j     = j0 + n;                    // output column 0..63
    const bool valid = (h == 0) && (j < OUT_W);  // lanes 16-31 are M=8 dups

    const float* img = data + b * (IMG_W * IMG_H);

    v8f c = {0.f, 0.f, 0.f, 0.f, 0.f, 0.f, 0.f, 0.f};
#pragma unroll
    for (int k = 0; k < 4; ++k) {
        // A: every row of A == w, so lane just needs its K slots.
        v2f a;
        a.x = w[4 * k + 2 * h + 0];
        a.y = w[4 * k + 2 * h + 1];
        // B: patch element (ki=k, kj=2h+{0,1}) of output column n.
        const float* rp = img + (row + k) * IMG_W;
        int col = j0 + n + 2 * h;
        col = col > 62 ? 62 : col;               // clamp (only garbage lanes)
        v2f bm;
        bm.x = rp[col];
        bm.y = rp[col + 1];
        // (neg_a, A, neg_b, B, c_mod, C, reuse_a, reuse_b)
        c = __builtin_amdgcn_wmma_f32_16x16x4_f32(
                false, a, false, bm, (short)0, c, false, false);
    }

    // D row M=0 lives in c[0], lanes 0-15 (N = lane). Lanes 16-31 hold the
    // identical M=8 row; mask them out via `valid`.
    float logit = c[0] + bptr[0];
    float s = valid ? (1.0f / (1.0f + expf(-logit))) : 0.0f;

    // wave32 reduction
#pragma unroll
    for (int off = 16; off > 0; off >>= 1)
        s += __shfl_xor(s, off, 32);
    if (lane == 0) wsum[wv] = s;
    __syncthreads();
    if (threadIdx.x == 0) {
        float bs = 0.0f;
#pragma unroll
        for (int i = 0; i < 8; ++i) bs += wsum[i];
        atomicAdd(acc, bs);
    }
}

// qexp is analytically 0 for this circuit (theta in {0,pi} on the uniform
// superposition only injects a global phase; CX permutes equal amplitudes;
// p1 = 0.5 exactly). Output[b] = 0.5*classical_mean + 0.5*0.
__global__ void finalize_kernel(const float* __restrict__ acc,
                                float* __restrict__ out) {
    int i = blockIdx.x * blockDim.x + threadIdx.x;
    if (i < BATCH) {
        float mean = acc[0] * (1.0f / (float)N_OUT_TOTAL);
        out[i] = 0.5f * mean;
    }
}

extern "C" void kernel_launch(void* const* d_in, const int* in_sizes, int n_in,
                              void* d_out, int out_size, void* d_ws, size_t ws_size,
                              hipStream_t stream) {
    const float* data = (const float*)d_in[0];   // (256,1,64,64) f32
    const float* cw   = (const float*)d_in[1];   // (1,1,4,4)     f32
    const float* cb   = (const float*)d_in[2];   // (1,)          f32
    float* out = (float*)d_out;                  // 256 f32
    float* acc = (float*)d_ws;                   // scratch scalar

    zero_acc_kernel<<<1, 64, 0, stream>>>(acc);

    const int total_waves = BATCH * OUT_H * 4;   // 62,464
    conv_sigmoid_reduce_wmma<<<total_waves / 8, 256, 0, stream>>>(data, cw, cb, acc);

    finalize_kernel<<<1, 256, 0, stream>>>(acc, out);
}